// Jaccard_14585708937908
// MI455X (gfx1250) — compile-verified
//
#include <hip/hip_runtime.h>
#include <hip/hip_bf16.h>

typedef __attribute__((ext_vector_type(2))) float v2f;
typedef __attribute__((ext_vector_type(4))) float v4f;
typedef __attribute__((ext_vector_type(8))) float v8f;

#define D_FEAT 128
#define THRESHOLD 0.01f
#define EPS 1e-8f

// ---------------------------------------------------------------------------
// Pass 1: node_norm[n] = sqrt(sum_k feat[n][k]^2).  One wave32 per node:
// lane loads float4 (row is 512B, fully coalesced), 5-step shfl_xor reduce.
// ---------------------------------------------------------------------------
__global__ __launch_bounds__(256) void jaccard_node_norm_kernel(
    const float* __restrict__ feat, float* __restrict__ norms, int n_nodes) {
  int wave = (int)((blockIdx.x * blockDim.x + threadIdx.x) >> 5);
  int lane = (int)(threadIdx.x & 31);
  if (wave >= n_nodes) return;  // wave-uniform exit

  const float* row = feat + (size_t)wave * D_FEAT + lane * 4;
  v4f v = *reinterpret_cast<const v4f*>(row);
  float s = v.x * v.x + v.y * v.y + v.z * v.z + v.w * v.w;
  s += __shfl_xor(s, 16, 32);
  s += __shfl_xor(s, 8, 32);
  s += __shfl_xor(s, 4, 32);
  s += __shfl_xor(s, 2, 32);
  s += __shfl_xor(s, 1, 32);
  if (lane == 0) norms[wave] = sqrtf(s);
}

// ---------------------------------------------------------------------------
// Pass 2: one wave32 per tile of 16 edges.  32 chained v_wmma_f32_16x16x4_f32
// accumulate A(src rows) x B(dst rows); the 16 per-edge dots are the diagonal.
//
// Per-lane operand layout (ISA 7.12.2, 32-bit 16x4 A; B mirrors with M->N):
//   m  = lane & 15        (edge within tile, used as both M and N index)
//   kb = 2*(lane >> 4)    (K sub-offset within each K=4 chunk)
//   VGPR0/1 of A = feat[src[m]][4c + kb + {0,1}]   -> one b64 load
//   VGPR0/1 of B = feat[dst[m]][4c + kb + {0,1}]   -> one b64 load
// ---------------------------------------------------------------------------
__global__ __launch_bounds__(256) void jaccard_edge_sim_kernel(
    const int* __restrict__ edge_index, const float* __restrict__ edge_weight,
    const float* __restrict__ feat, const float* __restrict__ norms,
    float* __restrict__ out, int E, int n_tiles) {
  int wave = (int)((blockIdx.x * blockDim.x + threadIdx.x) >> 5);
  int lane = (int)(threadIdx.x & 31);
  if (wave >= n_tiles) return;  // wave-uniform exit; EXEC all-1 below

  int m = lane & 15;
  int e = wave * 16 + m;
  bool valid = (e < E);
  int ec = valid ? e : 0;  // clamp so every lane issues loads (no divergence)

  int s = edge_index[ec];      // row 0 of [2,E]
  int d = edge_index[E + ec];  // row 1 of [2,E]

  int kb = (lane >> 4) << 1;
  const float* ap = feat + (size_t)s * D_FEAT + kb;
  const float* bp = feat + (size_t)d * D_FEAT + kb;

  v8f acc = {0.f, 0.f, 0.f, 0.f, 0.f, 0.f, 0.f, 0.f};
  for (int c0 = 0; c0 < 32; c0 += 8) {
#pragma unroll
    for (int u = 0; u < 8; ++u) {
      int c = c0 + u;
      v2f a = *reinterpret_cast<const v2f*>(ap + 4 * c);
      v2f b = *reinterpret_cast<const v2f*>(bp + 4 * c);
      // 8 args: (neg_a, A, neg_b, B, c_mod, C, reuse_a, reuse_b)
      acc = __builtin_amdgcn_wmma_f32_16x16x4_f32(
          false, a, false, b, (short)0, acc, false, false);
    }
  }

  // Diagonal extraction: D[i][i] is in acc VGPR (i&7), lane i (i<8) or 16+i.
  int r = m & 7;
  int srcLane = (m < 8) ? m : (m + 16);
  float inner = 0.f;
#pragma unroll
  for (int rr = 0; rr < 8; ++rr) {
    float t = __shfl(acc[rr], srcLane, 32);
    if (rr == r) inner = t;
  }

  if (lane < 16 && valid) {
    float ns = norms[s];
    float nd = norms[d];
    float sim = inner / (ns * nd + EPS);
    float keep = (sim >= THRESHOLD) ? 1.f : 0.f;
    float dd = (s == d) ? 2.f : 1.f;  // triu + triu^T doubles the diagonal
    out[e] = edge_weight[e] * keep * dd;
  }
}

extern "C" void kernel_launch(void* const* d_in, const int* in_sizes, int n_in,
                              void* d_out, int out_size, void* d_ws, size_t ws_size,
                              hipStream_t stream) {
  const int* edge_index = (const int*)d_in[0];     // [2, E] int32
  const float* edge_weight = (const float*)d_in[1];  // [E]
  const float* features = (const float*)d_in[2];     // [N, 128]
  float* out = (float*)d_out;
  float* norms = (float*)d_ws;  // n_nodes floats of scratch

  int E = in_sizes[1];
  int n_nodes = in_sizes[2] / D_FEAT;

  // Pass 1: norms (one wave per node, 8 waves / 256-thread block)
  int nblk = (n_nodes + 7) / 8;
  jaccard_node_norm_kernel<<<nblk, 256, 0, stream>>>(features, norms, n_nodes);

  // Pass 2: edge similarities (one wave per 16-edge tile)
  int n_tiles = (E + 15) / 16;
  int eblk = (n_tiles + 7) / 8;
  jaccard_edge_sim_kernel<<<eblk, 256, 0, stream>>>(
      edge_index, edge_weight, features, norms, out, E, n_tiles);
}